// VQCodebook_16587163697773
// MI455X (gfx1250) — compile-verified
//
#include <hip/hip_runtime.h>

typedef float v2f __attribute__((ext_vector_type(2)));
typedef float v8f __attribute__((ext_vector_type(8)));

#define SLOTS 512
#define DDIM  64
#define CBSTR 65   // padded codebook row stride (banks)
#define TSTR  17   // padded 16x16 tile stride
#define NWAVE 16   // waves per workgroup
#define NTHR  (NWAVE * 32)

__device__ __forceinline__ float fexp(float x) {
  return __builtin_amdgcn_exp2f(x * 1.4426950408889634f);
}
__device__ __forceinline__ float flog(float x) {
  return __builtin_amdgcn_logf(x) * 0.6931471805599453f;
}
// wave-internal LDS ordering: DS ops are in-order per wave on CDNA5; just stop
// the compiler from reordering around the store->load handoff.
#define WAVE_SYNC() do { __builtin_amdgcn_wave_barrier(); asm volatile("" ::: "memory"); } while (0)

__global__ void vq_init_scalars(float* kl, float* commit) {
  if (threadIdx.x == 0) { *kl = 0.f; *commit = 0.f; }
}

__global__ __launch_bounds__(NTHR, 1)
void vq_fused(const float* __restrict__ z_e, const float* __restrict__ cbg,
              const float* __restrict__ u, float* __restrict__ zq_out,
              float* __restrict__ idx_out, float* __restrict__ kl_out,
              float* __restrict__ commit_out) {
  extern __shared__ float smem[];
  float* cb  = smem;                         // 512*65
  float* cbn = cb + SLOTS * CBSTR;           // 512
  const int tid  = threadIdx.x;
  const int lane = tid & 31;
  const int wv   = tid >> 5;
  float* zw  = cbn + SLOTS + wv * 1600;      // 16*65 z tile
  float* t1  = zw + 16 * CBSTR;              // 16*17 logits tile
  float* t2  = t1 + 16 * TSTR;               // 16*17 gumbel-logits tile
  float* znl = t2 + 16 * TSTR;               // 16 row norms

  // ---- cooperative codebook load (padded) + slot norms ----
  for (int i = tid; i < SLOTS * DDIM; i += NTHR)
    cb[(i >> 6) * CBSTR + (i & 63)] = cbg[i];
  __syncthreads();
  for (int s = tid; s < SLOTS; s += NTHR) {
    float acc = 0.f;
    #pragma unroll 8
    for (int c = 0; c < DDIM; ++c) { float v = cb[s * CBSTR + c]; acc += v * v; }
    cbn[s] = acc;
  }
  __syncthreads();

  const int tile = blockIdx.x * NWAVE + wv;  // one 16-row tile per wave
  const int row0 = tile * 16;
  const int b    = row0 >> 12;
  const int x    = (row0 >> 6) & 63;
  const int y0   = row0 & 63;
  const long zbase = (long)b * (DDIM * 4096) + x * 64 + y0;

  // ---- z tile (16 rows x 64 dims) into LDS, row-major padded ----
  for (int i = lane; i < 16 * DDIM; i += 32) {
    int yy = i & 15, c = i >> 4;
    zw[yy * CBSTR + c] = z_e[zbase + (long)c * 4096 + yy];
  }
  WAVE_SYNC();
  if (lane < 16) {
    float acc = 0.f;
    for (int c = 0; c < DDIM; ++c) { float v = zw[lane * CBSTR + c]; acc += v * v; }
    znl[lane] = acc;
  }
  WAVE_SYNC();

  const int m    = lane & 15;
  const int hi   = (lane >> 4) & 1;
  const int koff = hi * 2;

  // online-softmax state (per row, held in lanes 0..15)
  float m1 = -__builtin_inff(), Z1 = 0.f, A1 = 0.f;  // plain softmax: max, sum, sum e*l
  float m2 = -__builtin_inff(), Z2 = 0.f;            // gumbel/T softmax
  float bestv = -__builtin_inff(); int besti = 0;
  float f2 = 0.f;
  v8f acc0 = {}, acc1 = {}, acc2 = {}, acc3 = {};    // z_q accum, 4 d-chunks of 16

  const float* urow = u + (long)row0 * SLOTS;        // 16 rows x 512 noise tile

  for (int sc = 0; sc < 32; ++sc) {
    const int sbase = sc * 16;

    // ---- issue u loads for this chunk first (latency hidden by WMMAs) ----
    float uu[8];
    #pragma unroll
    for (int r = 0; r < 8; ++r)
      uu[r] = urow[(r + hi * 8) * SLOTS + sbase + m];
    // prefetch next chunk's u tile (global_prefetch_b8): lane pairs cover rows
    if (sc < 31)
      __builtin_prefetch(&urow[(lane >> 1) * SLOTS + sbase + 16 + (lane & 1) * 8], 0, 1);

    // ---- distances dot tile (16 rows x 16 slots), K = 64 ----
    v8f dot = {};
    #pragma unroll
    for (int kb = 0; kb < DDIM; kb += 4) {
      v2f a, bb;
      a.x  = zw[m * CBSTR + kb + koff];
      a.y  = zw[m * CBSTR + kb + koff + 1];
      bb.x = cb[(sbase + m) * CBSTR + kb + koff];
      bb.y = cb[(sbase + m) * CBSTR + kb + koff + 1];
      dot = __builtin_amdgcn_wmma_f32_16x16x4_f32(false, a, false, bb, (short)0, dot, false, false);
    }
    // ---- logits + gumbel noise; stash tiles ----
    #pragma unroll
    for (int r = 0; r < 8; ++r) {
      int M = r + hi * 8;
      float l = 2.f * dot[r] - cbn[sbase + m] - znl[M];            // l = -distance
      float g = -flog(-flog(uu[r]));
      t1[M * TSTR + m] = l;
      t2[M * TSTR + m] = 2.f * (l + g);                            // (l+g)/T, T=0.5
    }
    WAVE_SYNC();
    // ---- per-row online stats ----
    if (lane < 16) {
      float m1n = m1, m2n = m2;
      #pragma unroll
      for (int s = 0; s < 16; ++s) {
        m1n = fmaxf(m1n, t1[lane * TSTR + s]);
        m2n = fmaxf(m2n, t2[lane * TSTR + s]);
      }
      float f1 = fexp(m1 - m1n); f2 = fexp(m2 - m2n);
      Z1 *= f1; A1 *= f1; Z2 *= f2;
      #pragma unroll
      for (int s = 0; s < 16; ++s) {
        float l  = t1[lane * TSTR + s];
        float e1 = fexp(l - m1n);
        Z1 += e1; A1 += e1 * l;
        float tv = t2[lane * TSTR + s];
        Z2 += fexp(tv - m2n);
        if (tv > bestv) { bestv = tv; besti = sbase + s; }
      }
      m1 = m1n; m2 = m2n;
    }
    // ---- rescale z_q accumulators by per-row factor ----
    #pragma unroll
    for (int r = 0; r < 8; ++r) {
      float f = __shfl(f2, r + hi * 8, 32);
      acc0[r] *= f; acc1[r] *= f; acc2[r] *= f; acc3[r] *= f;
    }
    float m2b = __shfl(m2, m, 32);
    // ---- z_q accumulation: A = exp weights, B = codebook rows ----
    #pragma unroll
    for (int kb = 0; kb < 16; kb += 4) {
      v2f aw;
      aw.x = fexp(t2[m * TSTR + kb + koff]     - m2b);
      aw.y = fexp(t2[m * TSTR + kb + koff + 1] - m2b);
      int s0 = (sbase + kb + koff) * CBSTR;
      int s1 = s0 + CBSTR;
      v2f b0, b1, b2, b3;
      b0.x = cb[s0 +  0 + m]; b0.y = cb[s1 +  0 + m];
      b1.x = cb[s0 + 16 + m]; b1.y = cb[s1 + 16 + m];
      b2.x = cb[s0 + 32 + m]; b2.y = cb[s1 + 32 + m];
      b3.x = cb[s0 + 48 + m]; b3.y = cb[s1 + 48 + m];
      acc0 = __builtin_amdgcn_wmma_f32_16x16x4_f32(false, aw, false, b0, (short)0, acc0, false, false);
      acc1 = __builtin_amdgcn_wmma_f32_16x16x4_f32(false, aw, false, b1, (short)0, acc1, false, false);
      acc2 = __builtin_amdgcn_wmma_f32_16x16x4_f32(false, aw, false, b2, (short)0, acc2, false, false);
      acc3 = __builtin_amdgcn_wmma_f32_16x16x4_f32(false, aw, false, b3, (short)0, acc3, false, false);
    }
    WAVE_SYNC();
  }

  // ---- scalar outputs per row ----
  float rz2 = 0.f;
  if (lane < 16) {
    rz2 = 1.f / Z2;
    float spl = A1 / Z1;                                   // sum p*logit
    float kl  = spl - m1 - flog(Z1) + 6.238324625039508f;  // + log(512)
    atomicAdd(kl_out, kl * (1.f / 32.f));
    atomicAdd(commit_out, -spl * (1.f / 32.f));            // sum p*dist = -sum p*l
    idx_out[row0 + lane] = (float)besti;
  }
  // ---- normalize and store z_q (transpose via LDS for coalescing) ----
  #pragma unroll
  for (int r = 0; r < 8; ++r) {
    float f = __shfl(rz2, r + hi * 8, 32);
    acc0[r] *= f; acc1[r] *= f; acc2[r] *= f; acc3[r] *= f;
  }
  #pragma unroll
  for (int dc = 0; dc < 4; ++dc) {
    v8f a = (dc == 0) ? acc0 : (dc == 1) ? acc1 : (dc == 2) ? acc2 : acc3;
    #pragma unroll
    for (int r = 0; r < 8; ++r)
      t1[(r + hi * 8) * TSTR + m] = a[r];                  // [row M][dcol N]
    WAVE_SYNC();
    #pragma unroll
    for (int r = 0; r < 8; ++r) {
      int dd = r + hi * 8;
      zq_out[zbase + (long)(dc * 16 + dd) * 4096 + m] = t1[m * TSTR + dd];
    }
    WAVE_SYNC();
  }
}

extern "C" void kernel_launch(void* const* d_in, const int* in_sizes, int n_in,
                              void* d_out, int out_size, void* d_ws, size_t ws_size,
                              hipStream_t stream) {
  const float* z_e = (const float*)d_in[0];
  const float* cbg = (const float*)d_in[1];
  const float* u   = (const float*)d_in[2];
  float* out    = (float*)d_out;
  float* zq     = out;                 // 32*64*64*64 = 524288
  float* idx    = out + 524288;        // 131072
  float* kl     = out + 655360;        // 1
  float* commit = out + 655361;        // 1

  const int smemBytes = (SLOTS * CBSTR + SLOTS + NWAVE * 1600) * 4;  // ~232 KB
  (void)hipFuncSetAttribute((const void*)vq_fused,
                            hipFuncAttributeMaxDynamicSharedMemorySize, smemBytes);

  vq_init_scalars<<<1, 1, 0, stream>>>(kl, commit);
  // 131072 rows / 16 per wave / 16 waves per block = 512 blocks
  vq_fused<<<512, NTHR, smemBytes, stream>>>(z_e, cbg, u, zq, idx, kl, commit);
}